// MultiHeadEGRETLayer_71725953843947
// MI455X (gfx1250) — compile-verified
//
#include <hip/hip_runtime.h>
#include <math.h>

#define NN   50000
#define EE   400000
#define IND  256
#define OUTD 64
#define EDD  64
#define HH   4
#define HO   (HH * OUTD)        // 256
#define ATTW (2 * OUTD + EDD)   // 192

typedef __attribute__((ext_vector_type(2))) float v2f;
typedef __attribute__((ext_vector_type(8))) float v8f;

// ---------------------------------------------------------------------------
// Prep: v_ea[h][f] = sum_g W_eatt[h][f][g] * We[h][g]
//       c_ea[h]    = sum_g b_eatt[h][g]   * We[h][g] + b_att[h]
// ---------------------------------------------------------------------------
__global__ void prep_kernel(const float* __restrict__ Watt,
                            const float* __restrict__ batt,
                            const float* __restrict__ Weatt,
                            const float* __restrict__ beatt,
                            float* __restrict__ v_ea,
                            float* __restrict__ c_ea) {
    int tid = threadIdx.x;            // 256 threads
    int h = tid >> 6, f = tid & 63;
    const float* we = Watt + h * ATTW + 2 * OUTD;       // We[h][*]
    const float* wr = Weatt + ((size_t)h * EDD + f) * EDD;
    float s = 0.f;
    for (int g = 0; g < EDD; ++g) s += wr[g] * we[g];
    v_ea[h * EDD + f] = s;
    if (f == 0) {
        float c = batt[h];
        const float* be = beatt + h * EDD;
        for (int g = 0; g < EDD; ++g) c += be[g] * we[g];
        c_ea[h] = c;
    }
}

// ---------------------------------------------------------------------------
// Init: zero the node-output region, seg_max = -inf, seg_sum = 0
// ---------------------------------------------------------------------------
__global__ void init_kernel(float* __restrict__ out,
                            float* __restrict__ seg_max,
                            float* __restrict__ seg_sum) {
    size_t tid = (size_t)blockIdx.x * blockDim.x + threadIdx.x;
    size_t stride = (size_t)gridDim.x * blockDim.x;
    for (size_t i = tid; i < (size_t)NN * HO; i += stride) out[i] = 0.f;
    for (size_t i = tid; i < (size_t)HH * NN; i += stride) {
        seg_max[i] = -INFINITY;
        seg_sum[i] = 0.f;
    }
}

// ---------------------------------------------------------------------------
// z = x @ W_fc + b_fc  via V_WMMA_F32_16X16X4_F32.
// Wave = one 16-row tile x one head (64 cols = 4 WMMA col-tiles).
// z layout: [n][h*64+o]  (row-major N x 256) -> final concat is trivial.
// ---------------------------------------------------------------------------
__global__ __launch_bounds__(128) void gemm_z_kernel(const float* __restrict__ x,
                                                     const float* __restrict__ Wfc,
                                                     const float* __restrict__ bfc,
                                                     float* __restrict__ z) {
    const int wave = threadIdx.x >> 5;
    const int lane = threadIdx.x & 31;
    const int rowTile = blockIdx.x * 4 + wave;      // 16-row tile id
    if (rowTile >= NN / 16) return;                  // wave-uniform guard
    const int h = blockIdx.y;                        // head = 64-col group
    const int m  = lane & 15;
    const int kb = (lane >> 4) << 1;                 // 0 or 2 (A/B K sub-slice)

    const float* xrow = x + (size_t)(rowTile * 16 + m) * IND;
    const float* wh   = Wfc + (size_t)h * IND * OUTD;

    v8f acc0 = {}, acc1 = {}, acc2 = {}, acc3 = {};
    for (int k0 = 0; k0 < IND; k0 += 4) {
        v2f a;                                       // A[m][k0+kb], A[m][k0+kb+1]
        a.x = xrow[k0 + kb];
        a.y = xrow[k0 + kb + 1];
        const float* w0 = wh + (size_t)(k0 + kb) * OUTD;
        const float* w1 = w0 + OUTD;
        v2f b;
        b.x = w0[0 * 16 + m]; b.y = w1[0 * 16 + m];
        acc0 = __builtin_amdgcn_wmma_f32_16x16x4_f32(false, a, false, b, (short)0, acc0, false, false);
        b.x = w0[1 * 16 + m]; b.y = w1[1 * 16 + m];
        acc1 = __builtin_amdgcn_wmma_f32_16x16x4_f32(false, a, false, b, (short)0, acc1, false, false);
        b.x = w0[2 * 16 + m]; b.y = w1[2 * 16 + m];
        acc2 = __builtin_amdgcn_wmma_f32_16x16x4_f32(false, a, false, b, (short)0, acc2, false, false);
        b.x = w0[3 * 16 + m]; b.y = w1[3 * 16 + m];
        acc3 = __builtin_amdgcn_wmma_f32_16x16x4_f32(false, a, false, b, (short)0, acc3, false, false);
    }
    // D layout: VGPR r -> row = r + 8*(lane>>4), col = lane&15 within tile
    const int rbase = rowTile * 16 + ((lane >> 4) << 3);
    #pragma unroll
    for (int r = 0; r < 8; ++r) {
        size_t row = (size_t)(rbase + r) * HO + h * OUTD;
        z[row + 0 * 16 + m] = acc0[r] + bfc[h * OUTD + 0 * 16 + m];
        z[row + 1 * 16 + m] = acc1[r] + bfc[h * OUTD + 1 * 16 + m];
        z[row + 2 * 16 + m] = acc2[r] + bfc[h * OUTD + 2 * 16 + m];
        z[row + 3 * 16 + m] = acc3[r] + bfc[h * OUTD + 3 * 16 + m];
    }
}

// ---------------------------------------------------------------------------
// a_i[h][n] = z[n, h*64:] . Wi[h] ;  a_j likewise with Wj
// ---------------------------------------------------------------------------
__global__ void att_node_kernel(const float* __restrict__ z,
                                const float* __restrict__ Watt,
                                float* __restrict__ a_i,
                                float* __restrict__ a_j) {
    int tid = blockIdx.x * blockDim.x + threadIdx.x;
    if (tid >= NN * HH) return;
    int n = tid >> 2;
    int h = tid & 3;
    const float* zr = z + (size_t)n * HO + h * OUTD;
    const float* wi = Watt + h * ATTW;
    const float* wj = wi + OUTD;
    float si = 0.f, sj = 0.f;
    for (int o = 0; o < OUTD; ++o) {
        float v = zr[o];
        si += v * wi[o];
        sj += v * wj[o];
    }
    a_i[h * NN + n] = si;
    a_j[h * NN + n] = sj;
}

// ---------------------------------------------------------------------------
// e[h][e] = leaky_relu(a_i[h][dst] + a_j[h][src] + attr.v_ea[h] + c_ea[h])
// ---------------------------------------------------------------------------
__global__ void edge_logits_kernel(const int* __restrict__ ei,
                                   const float* __restrict__ eattr,
                                   const float* __restrict__ v_ea,
                                   const float* __restrict__ c_ea,
                                   const float* __restrict__ a_i,
                                   const float* __restrict__ a_j,
                                   float* __restrict__ elog) {
    __shared__ float vsh[HH * EDD];
    __shared__ float csh[HH];
    for (int i = threadIdx.x; i < HH * EDD; i += blockDim.x) vsh[i] = v_ea[i];
    if (threadIdx.x < HH) csh[threadIdx.x] = c_ea[threadIdx.x];
    __syncthreads();
    int e = blockIdx.x * blockDim.x + threadIdx.x;
    if (e >= EE) return;
    int src = ei[e];
    int dst = ei[EE + e];
    const float* ar = eattr + (size_t)e * EDD;
    float s0 = 0.f, s1 = 0.f, s2 = 0.f, s3 = 0.f;
    for (int f = 0; f < EDD; ++f) {
        float v = ar[f];
        s0 += v * vsh[0 * EDD + f];
        s1 += v * vsh[1 * EDD + f];
        s2 += v * vsh[2 * EDD + f];
        s3 += v * vsh[3 * EDD + f];
    }
    float a0 = s0 + a_i[0 * NN + dst] + a_j[0 * NN + src] + csh[0];
    float a1 = s1 + a_i[1 * NN + dst] + a_j[1 * NN + src] + csh[1];
    float a2 = s2 + a_i[2 * NN + dst] + a_j[2 * NN + src] + csh[2];
    float a3 = s3 + a_i[3 * NN + dst] + a_j[3 * NN + src] + csh[3];
    elog[0 * EE + e] = a0 > 0.f ? a0 : 0.2f * a0;
    elog[1 * EE + e] = a1 > 0.f ? a1 : 0.2f * a1;
    elog[2 * EE + e] = a2 > 0.f ? a2 : 0.2f * a2;
    elog[3 * EE + e] = a3 > 0.f ? a3 : 0.2f * a3;
}

// Robust float atomic max (int punning for >=0, uint for <0)
__device__ __forceinline__ void atomicMaxFloat(float* addr, float value) {
    if (value >= 0.f) {
        atomicMax((int*)addr, __float_as_int(value));
    } else {
        atomicMin((unsigned int*)addr, __float_as_uint(value));
    }
}

__global__ void segmax_kernel(const int* __restrict__ ei,
                              const float* __restrict__ elog,
                              float* __restrict__ seg_max) {
    int tid = blockIdx.x * blockDim.x + threadIdx.x;
    if (tid >= HH * EE) return;
    int h = tid / EE;
    int e = tid - h * EE;
    int dst = ei[EE + e];
    atomicMaxFloat(&seg_max[h * NN + dst], elog[tid]);
}

__global__ void expsum_kernel(const int* __restrict__ ei,
                              float* __restrict__ elog,
                              const float* __restrict__ seg_max,
                              float* __restrict__ seg_sum) {
    int tid = blockIdx.x * blockDim.x + threadIdx.x;
    if (tid >= HH * EE) return;
    int h = tid / EE;
    int e = tid - h * EE;
    int dst = ei[EE + e];
    float ex = __expf(elog[tid] - seg_max[h * NN + dst]);
    elog[tid] = ex;
    atomicAdd(&seg_sum[h * NN + dst], ex);
}

__global__ void alpha_kernel(const int* __restrict__ ei,
                             float* __restrict__ elog,       // holds exp_e, becomes alpha
                             const float* __restrict__ seg_sum,
                             float* __restrict__ out_alpha) {
    int tid = blockIdx.x * blockDim.x + threadIdx.x;
    if (tid >= HH * EE) return;
    int h = tid / EE;
    int e = tid - h * EE;
    int dst = ei[EE + e];
    float al = elog[tid] / seg_sum[h * NN + dst];
    elog[tid] = al;
    out_alpha[tid] = al;    // output layout [h][e] == flat tid
}

// ---------------------------------------------------------------------------
// Scatter: out[dst, h*64+o] += alpha[h][e] * (z[src, h*64+o] + ez[h][e][o])
// ez fused: ez = edge_attr[e] @ W_edge[h] + b_edge[h]  (W_edge[h] in LDS, 16KB)
// Block = 256 thr = 4 edge slots x 64 outs; grid.y = head
// ---------------------------------------------------------------------------
__global__ __launch_bounds__(256) void scatter_kernel(const int* __restrict__ ei,
                                                      const float* __restrict__ eattr,
                                                      const float* __restrict__ z,
                                                      const float* __restrict__ Wedge,
                                                      const float* __restrict__ bedge,
                                                      const float* __restrict__ alpha,
                                                      float* __restrict__ out) {
    const int h = blockIdx.y;
    __shared__ float wsh[EDD * OUTD];       // 16 KB: W_edge[h]
    __shared__ float ash[4][EDD];
    for (int i = threadIdx.x; i < EDD * OUTD; i += blockDim.x)
        wsh[i] = Wedge[(size_t)h * EDD * OUTD + i];
    __syncthreads();

    const int slot = threadIdx.x >> 6;      // 0..3
    const int o    = threadIdx.x & 63;
    const float bo = bedge[h * OUTD + o];
    const int nbatch = (EE + 3) / 4;

    for (int b = blockIdx.x; b < nbatch; b += gridDim.x) {
        int e = b * 4 + slot;
        __syncthreads();
        if (e < EE) ash[slot][o] = eattr[(size_t)e * EDD + o];
        __syncthreads();
        if (e >= EE) continue;
        int src = ei[e];
        int dst = ei[EE + e];
        float al = alpha[(size_t)h * EE + e];
        float ez = bo;
        #pragma unroll 8
        for (int f = 0; f < EDD; ++f) ez += ash[slot][f] * wsh[f * OUTD + o];
        float msg = al * (z[(size_t)src * HO + h * OUTD + o] + ez);
        atomicAdd(&out[(size_t)dst * HO + h * OUTD + o], msg);
    }
}

// ---------------------------------------------------------------------------
extern "C" void kernel_launch(void* const* d_in, const int* in_sizes, int n_in,
                              void* d_out, int out_size, void* d_ws, size_t ws_size,
                              hipStream_t stream) {
    const float* x     = (const float*)d_in[0];
    const int*   ei    = (const int*)  d_in[1];
    const float* eattr = (const float*)d_in[2];
    const float* Wfc   = (const float*)d_in[3];
    const float* bfc   = (const float*)d_in[4];
    const float* Watt  = (const float*)d_in[5];
    const float* batt  = (const float*)d_in[6];
    const float* Wedge = (const float*)d_in[7];
    const float* bedge = (const float*)d_in[8];
    const float* Weatt = (const float*)d_in[9];
    const float* beatt = (const float*)d_in[10];

    float* out       = (float*)d_out;                 // [N][H*OUT] then [H][E]
    float* out_alpha = out + (size_t)NN * HO;

    float* ws = (float*)d_ws;
    size_t off = 0;
    float* z       = ws + off; off += (size_t)NN * HO;   // 12.8M
    float* a_i     = ws + off; off += (size_t)HH * NN;
    float* a_j     = ws + off; off += (size_t)HH * NN;
    float* v_ea    = ws + off; off += HH * EDD;
    float* c_ea    = ws + off; off += 64;
    float* elog    = ws + off; off += (size_t)HH * EE;   // logits -> exp -> alpha
    float* seg_max = ws + off; off += (size_t)HH * NN;
    float* seg_sum = ws + off; off += (size_t)HH * NN;

    prep_kernel<<<1, 256, 0, stream>>>(Watt, batt, Weatt, beatt, v_ea, c_ea);
    init_kernel<<<2048, 256, 0, stream>>>(out, seg_max, seg_sum);

    dim3 ggrid((NN / 16 + 3) / 4, HH);                   // 782 x 4
    gemm_z_kernel<<<ggrid, 128, 0, stream>>>(x, Wfc, bfc, z);

    att_node_kernel<<<(NN * HH + 255) / 256, 256, 0, stream>>>(z, Watt, a_i, a_j);
    edge_logits_kernel<<<(EE + 255) / 256, 256, 0, stream>>>(ei, eattr, v_ea, c_ea, a_i, a_j, elog);
    segmax_kernel<<<(HH * EE + 255) / 256, 256, 0, stream>>>(ei, elog, seg_max);
    expsum_kernel<<<(HH * EE + 255) / 256, 256, 0, stream>>>(ei, elog, seg_max, seg_sum);
    alpha_kernel<<<(HH * EE + 255) / 256, 256, 0, stream>>>(ei, elog, seg_sum, out_alpha);

    dim3 sgrid(4096, HH);
    scatter_kernel<<<sgrid, 256, 0, stream>>>(ei, eattr, z, Wedge, bedge, elog, out);
}